// MultiBoxLoss_80822694576222
// MI455X (gfx1250) — compile-verified
//
#include <hip/hip_runtime.h>

static constexpr int kB  = 64;
static constexpr int kP  = 8732;
static constexpr int kC  = 81;
static constexpr int kO  = 32;
static constexpr int kBP = kB * kP;   // 558848 = 32 * 17464 -> all waves full

typedef float v2f __attribute__((ext_vector_type(2)));
typedef float v8f __attribute__((ext_vector_type(8)));

// Exact 32-lane sum via V_WMMA_F32_16X16X4_F32 chains. Result broadcast to all
// lanes. Requires EXEC all-ones (call only from fully-converged full waves).
__device__ __forceinline__ float wave_reduce_sum_wmma(float v) {
  v2f a;    a[0] = v;    a[1] = 0.0f;     // A[m,k]: v in one K slot per row
  v2f ones; ones[0] = 1.0f; ones[1] = 1.0f;
  v8f z = {0.f, 0.f, 0.f, 0.f, 0.f, 0.f, 0.f, 0.f};
  // D1[m,n] = v[m] + v[m+16]  (row sums of A, replicated over n)
  v8f d1 = __builtin_amdgcn_wmma_f32_16x16x4_f32(false, a, false, ones,
                                                 (short)0, z, false, false);
  // ones(16x4) x (pairs of D1 regs as 4x16 B): each acc element accumulates
  // four distinct s-values per WMMA; after 4 WMMAs acc = sum of all 32 lanes.
  v8f acc = z;
  v2f b; 
  b[0] = d1[0]; b[1] = d1[1];
  acc = __builtin_amdgcn_wmma_f32_16x16x4_f32(false, ones, false, b, (short)0, acc, false, false);
  b[0] = d1[2]; b[1] = d1[3];
  acc = __builtin_amdgcn_wmma_f32_16x16x4_f32(false, ones, false, b, (short)0, acc, false, false);
  b[0] = d1[4]; b[1] = d1[5];
  acc = __builtin_amdgcn_wmma_f32_16x16x4_f32(false, ones, false, b, (short)0, acc, false, false);
  b[0] = d1[6]; b[1] = d1[7];
  acc = __builtin_amdgcn_wmma_f32_16x16x4_f32(false, ones, false, b, (short)0, acc, false, false);
  return acc[0];
}

__global__ void k_init(int* npos, float* sums) {
  int t = threadIdx.x;
  if (t < kB) npos[t] = 0;
  if (t == 0) { sums[0] = 0.0f; sums[1] = 0.0f; }
}

// One block per batch: per-prior best truth (pack -> match[]), per-truth best
// prior (LDS atomicMax on packed key -> bpidx[]).
__global__ void k_match(const float* __restrict__ truths,
                        const float* __restrict__ priors,
                        int* __restrict__ match, int* __restrict__ bpidx) {
  const int b = blockIdx.x;
  const int tid = threadIdx.x;
  __shared__ float4 s_t[kO];
  __shared__ float  s_at[kO];
  __shared__ unsigned long long s_best[kO];
  if (tid < kO) {
    float4 t = ((const float4*)truths)[b * kO + tid];
    s_t[tid] = t;
    s_at[tid] = (t.z - t.x) * (t.w - t.y);
    s_best[tid] = 0ull;
  }
  __syncthreads();

  float bv[kO]; int bp[kO];
#pragma unroll
  for (int o = 0; o < kO; ++o) { bv[o] = 0.0f; bp[o] = 0; }

  for (int p = tid; p < kP; p += blockDim.x) {
    float4 pr = ((const float4*)priors)[p];
    float px1 = pr.x - pr.z * 0.5f, py1 = pr.y - pr.w * 0.5f;
    float px2 = pr.x + pr.z * 0.5f, py2 = pr.y + pr.w * 0.5f;
    float ap = (px2 - px1) * (py2 - py1);
    float bov = -1.0f; int boidx = 0;
#pragma unroll
    for (int o = 0; o < kO; ++o) {
      float4 t = s_t[o];
      float lx = fmaxf(t.x, px1), ly = fmaxf(t.y, py1);
      float rx = fminf(t.z, px2), ry = fminf(t.w, py2);
      float w = fmaxf(rx - lx, 0.0f), h = fmaxf(ry - ly, 0.0f);
      float inter = w * h;
      float iou = inter / (s_at[o] + ap - inter);
      if (iou > bov) { bov = iou; boidx = o; }        // first index on ties
      if (iou > bv[o]) { bv[o] = iou; bp[o] = p; }    // smallest p on ties
    }
    match[b * kP + p] = (bov >= 0.5f ? 0x100 : 0) | boidx;
  }
#pragma unroll
  for (int o = 0; o < kO; ++o) {
    unsigned long long key =
        ((unsigned long long)__float_as_uint(bv[o]) << 32) |
        (unsigned long long)(0xFFFFFFFFu - (unsigned)bp[o]);
    atomicMax(&s_best[o], key);
  }
  __syncthreads();
  if (tid < kO) {
    unsigned inv = (unsigned)(s_best[tid] & 0xFFFFFFFFull);
    bpidx[b * kO + tid] = (int)(0xFFFFFFFFu - inv);
  }
}

// Force every truth to keep its best prior (deterministic last-o-wins).
__global__ void k_override(const int* __restrict__ bpidx, int* __restrict__ match) {
  int b = threadIdx.x;
  if (b >= kB) return;
  for (int o = 0; o < kO; ++o) {
    int p = bpidx[b * kO + o];
    match[b * kP + p] = 0x100 | o;
  }
}

// conf_t finalize + smooth-L1 loc loss over positives + per-batch pos counts.
__global__ void k_finalize(const float* __restrict__ loc,
                           const float* __restrict__ truths,
                           const float* __restrict__ priors,
                           const int* __restrict__ labels,
                           int* __restrict__ match,
                           int* __restrict__ npos, float* __restrict__ sums) {
  int t = blockIdx.x * blockDim.x + threadIdx.x;   // grid sized to exactly kBP
  int b = t / kP, p = t - b * kP;
  int pk = match[t];
  int o = pk & 0xFF;
  bool pos = (pk & 0x100) != 0;
  int conf = pos ? (labels[b * kO + o] + 1) : 0;
  match[t] = conf;
  float sl = 0.0f;
  if (pos) {
    float4 tr = ((const float4*)truths)[b * kO + o];
    float4 pr = ((const float4*)priors)[p];
    float gcx = ((tr.x + tr.z) * 0.5f - pr.x) / (0.1f * pr.z);
    float gcy = ((tr.y + tr.w) * 0.5f - pr.y) / (0.1f * pr.w);
    float gw  = __logf((tr.z - tr.x) / pr.z) * 5.0f;   // / VAR_SIZE
    float gh  = __logf((tr.w - tr.y) / pr.w) * 5.0f;
    float4 l = ((const float4*)loc)[t];
    float d0 = fabsf(l.x - gcx), d1 = fabsf(l.y - gcy);
    float d2 = fabsf(l.z - gw),  d3 = fabsf(l.w - gh);
    sl  = (d0 < 1.f ? 0.5f * d0 * d0 : d0 - 0.5f);
    sl += (d1 < 1.f ? 0.5f * d1 * d1 : d1 - 0.5f);
    sl += (d2 < 1.f ? 0.5f * d2 * d2 : d2 - 0.5f);
    sl += (d3 < 1.f ? 0.5f * d3 * d3 : d3 - 0.5f);
    atomicAdd(&npos[b], 1);   // per-lane: waves may straddle batch boundary
  }
  float wsum = wave_reduce_sum_wmma(sl);             // EXEC all-ones here
  if ((threadIdx.x & 31) == 0) atomicAdd(&sums[0], wsum);
}

// Streaming CE pass: one wave per 81-class row (coalesced). Stores ce, and
// accumulates positive-row CE (positives are always selected).
__global__ void k_ce(const float* __restrict__ conf, const int* __restrict__ conf_t,
                     float* __restrict__ ce, float* __restrict__ sums) {
  int lane = threadIdx.x & 31;
  int wave = blockIdx.x * (blockDim.x >> 5) + (threadIdx.x >> 5);
  int nw = gridDim.x * (blockDim.x >> 5);
  for (int row = wave; row < kBP; row += nw) {
    const float* x = conf + (size_t)row * kC;
    float x0 = x[lane];
    float x1 = x[lane + 32];
    bool has2 = lane < (kC - 64);                    // lanes 0..16
    float x2 = has2 ? x[lane + 64] : -3.0e38f;
    float m = fmaxf(fmaxf(x0, x1), x2);
#pragma unroll
    for (int off = 16; off >= 1; off >>= 1) m = fmaxf(m, __shfl_xor(m, off, 32));
    float e = __expf(x0 - m) + __expf(x1 - m) + (has2 ? __expf(x2 - m) : 0.0f);
    float s = wave_reduce_sum_wmma(e);               // full wave, EXEC all-ones
    int tc = conf_t[row];
    float xt = x[tc];
    if (lane == 0) {
      float cv = __logf(s) + m - xt;
      ce[row] = cv;
      if (tc > 0) atomicAdd(&sums[1], cv);
    }
  }
}

// Per-batch exact top-k threshold (3-level radix histogram on CE bits, CE>=0
// so bits are order-isomorphic) + sum of selected-negative CE.
__global__ __launch_bounds__(1024) void k_select(const float* __restrict__ ce,
                                                 const int* __restrict__ conf_t,
                                                 const int* __restrict__ npos,
                                                 float* __restrict__ sums) {
  const int b = blockIdx.x, tid = threadIdx.x;
  __shared__ float    s_ce[kP];
  __shared__ unsigned s_hist[2048];
  __shared__ unsigned s_T, s_bin1, s_bin2;
  __shared__ int s_rem, s_none;
  __shared__ float s_w[32];

  int np = npos[b];
  int want = np * 3; if (want > kP) want = kP;

  for (int p = tid; p < kP; p += 1024) s_ce[p] = ce[b * kP + p];
  for (int i = tid; i < 2048; i += 1024) s_hist[i] = 0;
  __syncthreads();
  for (int p = tid; p < kP; p += 1024)
    atomicAdd(&s_hist[__float_as_uint(s_ce[p]) >> 20], 1u);
  __syncthreads();
  if (tid == 0) {
    if (want <= 0) { s_none = 1; s_T = 0xFFFFFFFFu; s_rem = 0; }
    else {
      s_none = 0; int cum = 0; unsigned t1 = 0; int rem = want;
      for (int i = 2047; i >= 0; --i) {
        int c = (int)s_hist[i];
        if (cum + c >= want) { t1 = (unsigned)i; rem = want - cum; break; }
        cum += c;
      }
      s_bin1 = t1; s_rem = rem;
    }
  }
  __syncthreads();

  if (!s_none) {
    unsigned t1 = s_bin1; int want2 = s_rem;
    for (int i = tid; i < 2048; i += 1024) s_hist[i] = 0;
    __syncthreads();
    for (int p = tid; p < kP; p += 1024) {
      unsigned bits = __float_as_uint(s_ce[p]);
      if ((bits >> 20) == t1) atomicAdd(&s_hist[(bits >> 9) & 0x7FFu], 1u);
    }
    __syncthreads();
    if (tid == 0) {
      int cum = 0; unsigned t2 = 0; int rem = want2;
      for (int i = 2047; i >= 0; --i) {
        int c = (int)s_hist[i];
        if (cum + c >= want2) { t2 = (unsigned)i; rem = want2 - cum; break; }
        cum += c;
      }
      s_bin2 = (t1 << 11) | t2; s_rem = rem;
    }
    __syncthreads();
    unsigned pre = s_bin2; int want3 = s_rem;
    for (int i = tid; i < 512; i += 1024) s_hist[i] = 0;
    __syncthreads();
    for (int p = tid; p < kP; p += 1024) {
      unsigned bits = __float_as_uint(s_ce[p]);
      if ((bits >> 9) == pre) atomicAdd(&s_hist[bits & 0x1FFu], 1u);
    }
    __syncthreads();
    if (tid == 0) {
      int cum = 0; unsigned t3 = 0; int rem = want3;
      for (int i = 511; i >= 0; --i) {
        int c = (int)s_hist[i];
        if (cum + c >= want3) { t3 = (unsigned)i; rem = want3 - cum; break; }
        cum += c;
      }
      s_T = (pre << 9) | t3; s_rem = rem;
    }
    __syncthreads();
  }

  const int none = s_none; const unsigned T = s_T; const int r = s_rem;
  float part = 0.0f;
  if (!none) {
    for (int p = tid; p < kP; p += 1024) {
      unsigned bits = __float_as_uint(s_ce[p]);
      bool sel = bits > T;
      if (!sel && bits == T) {           // index-order tie break (rare)
        int eq = 0;
        for (int q = 0; q < p; ++q) if (__float_as_uint(s_ce[q]) == T) ++eq;
        sel = eq < r;
      }
      if (sel && conf_t[b * kP + p] <= 0) part += s_ce[p];  // positives added in k_ce
    }
  }
  float wsum = wave_reduce_sum_wmma(part);             // reconverged, full waves
  if ((tid & 31) == 0) s_w[tid >> 5] = wsum;
  __syncthreads();
  if (tid < 32) {                                      // wave 0, EXEC all-ones
    float tot = wave_reduce_sum_wmma(s_w[tid]);
    if (tid == 0) atomicAdd(&sums[1], tot);
  }
}

__global__ void k_final(const int* __restrict__ npos, const float* __restrict__ sums,
                        float* __restrict__ out) {
  if (threadIdx.x == 0) {
    int n = 0;
    for (int b = 0; b < kB; ++b) n += npos[b];
    float N = (float)n;
    out[0] = sums[0] / N;
    out[1] = sums[1] / N;
  }
}

extern "C" void kernel_launch(void* const* d_in, const int* in_sizes, int n_in,
                              void* d_out, int out_size, void* d_ws, size_t ws_size,
                              hipStream_t stream) {
  (void)in_sizes; (void)n_in; (void)out_size; (void)ws_size;
  const float* loc    = (const float*)d_in[0];
  const float* conf   = (const float*)d_in[1];
  const float* priors = (const float*)d_in[2];
  const float* truths = (const float*)d_in[3];
  const int*   labels = (const int*)d_in[4];
  float* out = (float*)d_out;

  char* ws = (char*)d_ws;
  int*   match = (int*)ws;                                   // [kBP]
  float* ce    = (float*)(ws + (size_t)kBP * 4);             // [kBP]
  int*   bpidx = (int*)(ws + (size_t)kBP * 8);               // [kB*kO]
  int*   npos  = (int*)(ws + (size_t)kBP * 8 + (size_t)kB * kO * 4);   // [kB]
  float* sums  = (float*)(ws + (size_t)kBP * 8 + (size_t)kB * kO * 4 + (size_t)kB * 4); // [2]

  k_init    <<<1, 64, 0, stream>>>(npos, sums);
  k_match   <<<kB, 256, 0, stream>>>(truths, priors, match, bpidx);
  k_override<<<1, 64, 0, stream>>>(bpidx, match);
  k_finalize<<<kBP / 256, 256, 0, stream>>>(loc, truths, priors, labels, match, npos, sums);
  k_ce      <<<1024, 256, 0, stream>>>(conf, match, ce, sums);
  k_select  <<<kB, 1024, 0, stream>>>(ce, match, npos, sums);
  k_final   <<<1, 1, 0, stream>>>(npos, sums, out);
}